// GATLayer_70875550318711
// MI455X (gfx1250) — compile-verified
//
#include <hip/hip_runtime.h>
#include <hip/hip_bf16.h>

#define NN  50000
#define EE  800000
#define IND 128
#define HH  4
#define CC  32
#define HCC 128   // H*C

typedef __attribute__((ext_vector_type(2))) float v2f;
typedef __attribute__((ext_vector_type(8))) float v8f;

// ---------- order-preserving float <-> uint map for atomicMax over floats ----------
__device__ __forceinline__ unsigned fmap(float f) {
    unsigned u = __float_as_uint(f);
    return (u & 0x80000000u) ? ~u : (u | 0x80000000u);
}
__device__ __forceinline__ float funmap(unsigned u) {
    return __uint_as_float((u & 0x80000000u) ? (u & 0x7FFFFFFFu) : ~u);
}

// ---------- init workspace ----------
__global__ void k_init(float* out_pre, float* denom, unsigned* amax_u) {
    int i = blockIdx.x * blockDim.x + threadIdx.x;
    if (i < NN * HCC) out_pre[i] = 0.0f;
    if (i < NN * HH) { denom[i] = 0.0f; amax_u[i] = 0u; }  // 0u < fmap(any finite)
}

// ---------- dual GEMM: x_l = x@W_l + b_l ; x_r = x@W_r + b_r  (fp32 WMMA) ----------
// One block = 8 waves = one 16-row M tile. LDS-staged A tile shared by all 8 N tiles
// and both weight matrices. Each wave: 32 K-steps x 2 WMMAs = 64 v_wmma_f32_16x16x4_f32.
__global__ __launch_bounds__(256) void k_gemm_xl_xr(
    const float* __restrict__ x,
    const float* __restrict__ Wl, const float* __restrict__ bl,
    const float* __restrict__ Wr, const float* __restrict__ br,
    float* __restrict__ xl, float* __restrict__ xr)
{
    __shared__ float As[16][132];   // pad 128 -> 132 to break 64-bank conflicts
    const int tid  = threadIdx.x;
    const int row0 = blockIdx.x * 16;

    // cooperative load: 16 rows x 128 cols, float4-wide
    for (int i = tid; i < 16 * 32; i += 256) {
        int r  = i >> 5;
        int c4 = (i & 31) << 2;
        int gr = row0 + r;
        float4 v = make_float4(0.f, 0.f, 0.f, 0.f);
        if (gr < NN) v = *reinterpret_cast<const float4*>(&x[(size_t)gr * IND + c4]);
        As[r][c4 + 0] = v.x; As[r][c4 + 1] = v.y;
        As[r][c4 + 2] = v.z; As[r][c4 + 3] = v.w;
    }
    __syncthreads();

    const int lane = tid & 31;
    const int wave = tid >> 5;      // N tile 0..7
    const int half = lane >> 4;     // 0: lanes 0-15, 1: lanes 16-31
    const int lq   = lane & 15;
    const int col  = wave * 16 + lq;

    v8f accL = {0.f,0.f,0.f,0.f,0.f,0.f,0.f,0.f};
    v8f accR = {0.f,0.f,0.f,0.f,0.f,0.f,0.f,0.f};

    for (int k0 = 0; k0 < IND; k0 += 4) {
        // A 16x4 layout: lane holds row M=lq; VGPR0 -> K=k0+2*half, VGPR1 -> K+1
        const int ka = k0 + 2 * half;
        v2f a;  a.x  = As[lq][ka];
                a.y  = As[lq][ka + 1];
        // B 4x16 layout (mirror): VGPR0 -> K=k0+2*half row, lanes give N=col
        v2f bL; bL.x = Wl[(size_t)ka * HCC + col];
                bL.y = Wl[(size_t)(ka + 1) * HCC + col];
        v2f bR; bR.x = Wr[(size_t)ka * HCC + col];
                bR.y = Wr[(size_t)(ka + 1) * HCC + col];

        accL = __builtin_amdgcn_wmma_f32_16x16x4_f32(
                   false, a, false, bL, (short)0, accL, false, false);
        accR = __builtin_amdgcn_wmma_f32_16x16x4_f32(
                   false, a, false, bR, (short)0, accR, false, false);
    }

    const float bLc = bl[col];
    const float bRc = br[col];
    // D 16x16 layout: VGPR v -> row M = v + 8*half, lanes give N=col
    #pragma unroll
    for (int v = 0; v < 8; ++v) {
        int r = row0 + v + 8 * half;
        if (r < NN) {
            xl[(size_t)r * HCC + col] = accL[v] + bLc;
            xr[(size_t)r * HCC + col] = accR[v] + bRc;
        }
    }
}

// ---------- attention logits + segment max ----------
// one wave per (edge, head); lanes = channels (C==32==warpSize)
__global__ __launch_bounds__(256) void k_edge_logits(
    const int* __restrict__ src, const int* __restrict__ dst,
    const float* __restrict__ edge_attr,
    const float* __restrict__ We, const float* __restrict__ att,
    const float* __restrict__ xl, const float* __restrict__ xr,
    float* __restrict__ alpha, unsigned* __restrict__ amax_u)
{
    int gw   = (blockIdx.x * blockDim.x + threadIdx.x) >> 5;
    int lane = threadIdx.x & 31;
    int e = gw >> 2;
    int h = gw & 3;
    if (e >= EE) return;

    int s = src[e], d = dst[e];
    int hc = h * CC + lane;
    float v = xl[(size_t)s * HCC + hc] + xr[(size_t)d * HCC + hc]
            + edge_attr[e] * We[hc];
    v = (v > 0.f) ? v : 0.2f * v;              // LeakyReLU(0.2)
    float p = v * att[hc];
    #pragma unroll
    for (int o = 16; o > 0; o >>= 1) p += __shfl_xor(p, o, 32);
    if (lane == 0) {
        alpha[(size_t)e * HH + h] = p;
        atomicMax(&amax_u[(size_t)d * HH + h], fmap(p));
    }
}

// ---------- exp(alpha - amax) and segment denominator ----------
__global__ void k_softmax_num(const int* __restrict__ dst,
                              const unsigned* __restrict__ amax_u,
                              float* __restrict__ alpha,
                              float* __restrict__ denom)
{
    int i = blockIdx.x * blockDim.x + threadIdx.x;   // over E*H
    if (i >= EE * HH) return;
    int e = i >> 2, h = i & 3;
    int d = dst[e];
    float ex = expf(alpha[i] - funmap(amax_u[(size_t)d * HH + h]));
    alpha[i] = ex;
    atomicAdd(&denom[(size_t)d * HH + h], ex);
}

// ---------- normalize alpha once denom is complete (E*H divides, not E*HC) ----------
__global__ void k_norm_alpha(const int* __restrict__ dst,
                             const float* __restrict__ denom,
                             float* __restrict__ alpha)
{
    int i = blockIdx.x * blockDim.x + threadIdx.x;   // over E*H
    if (i >= EE * HH) return;
    int e = i >> 2, h = i & 3;
    int d = dst[e];
    alpha[i] = alpha[i] / (denom[(size_t)d * HH + h] + 1e-16f);
}

// ---------- weighted scatter-add aggregation (pure FMA + atomicAdd) ----------
__global__ void k_aggregate(const int* __restrict__ src, const int* __restrict__ dst,
                            const float* __restrict__ xl,
                            const float* __restrict__ alpha,
                            float* __restrict__ out_pre)
{
    long long i = (long long)blockIdx.x * blockDim.x + threadIdx.x;  // over E*HC
    if (i >= (long long)EE * HCC) return;
    int e  = (int)(i >> 7);
    int hc = (int)(i & 127);
    int h  = hc >> 5;
    int s = src[e], d = dst[e];
    float w = alpha[(size_t)e * HH + h];
    atomicAdd(&out_pre[(size_t)d * HCC + hc], xl[(size_t)s * HCC + hc] * w);
}

// ---------- BatchNorm statistics (one block per channel) ----------
__global__ __launch_bounds__(256) void k_bn_stats(const float* __restrict__ out_pre,
                                                  const float* __restrict__ bias,
                                                  float* __restrict__ mean,
                                                  float* __restrict__ var)
{
    int c = blockIdx.x;
    float b = bias[c];
    float s = 0.f, s2 = 0.f;
    for (int n = threadIdx.x; n < NN; n += 256) {
        float v = out_pre[(size_t)n * HCC + c] + b;
        s += v; s2 += v * v;
    }
    __shared__ float sh0[256], sh1[256];
    sh0[threadIdx.x] = s; sh1[threadIdx.x] = s2;
    __syncthreads();
    for (int o = 128; o > 0; o >>= 1) {
        if (threadIdx.x < o) {
            sh0[threadIdx.x] += sh0[threadIdx.x + o];
            sh1[threadIdx.x] += sh1[threadIdx.x + o];
        }
        __syncthreads();
    }
    if (threadIdx.x == 0) {
        float m = sh0[0] / (float)NN;
        mean[c] = m;
        var[c]  = sh1[0] / (float)NN - m * m;
    }
}

// ---------- BN affine + ELU ----------
__global__ void k_bn_elu(const float* __restrict__ out_pre,
                         const float* __restrict__ bias,
                         const float* __restrict__ mean, const float* __restrict__ var,
                         const float* __restrict__ gamma, const float* __restrict__ beta,
                         float* __restrict__ out)
{
    int i = blockIdx.x * blockDim.x + threadIdx.x;   // over N*HC
    if (i >= NN * HCC) return;
    int c = i & 127;
    float v = out_pre[i] + bias[c];
    v = (v - mean[c]) * rsqrtf(var[c] + 1e-5f) * gamma[c] + beta[c];
    out[i] = (v > 0.f) ? v : expm1f(v);
}

extern "C" void kernel_launch(void* const* d_in, const int* in_sizes, int n_in,
                              void* d_out, int out_size, void* d_ws, size_t ws_size,
                              hipStream_t stream) {
    const float* x         = (const float*)d_in[0];
    const int*   ei        = (const int*)  d_in[1];   // [2, E]
    const float* edge_attr = (const float*)d_in[2];
    const float* W_l       = (const float*)d_in[3];
    const float* b_l       = (const float*)d_in[4];
    const float* W_r       = (const float*)d_in[5];
    const float* b_r       = (const float*)d_in[6];
    const float* W_e       = (const float*)d_in[7];
    const float* att       = (const float*)d_in[8];
    const float* bias      = (const float*)d_in[9];
    const float* gamma     = (const float*)d_in[10];
    const float* beta      = (const float*)d_in[11];
    float* out = (float*)d_out;

    const int* src = ei;
    const int* dst = ei + EE;

    // workspace carve-up (floats)
    float*    ws      = (float*)d_ws;
    float*    xl      = ws;                         // N*HC
    float*    xr      = xl + (size_t)NN * HCC;      // N*HC
    float*    alpha   = xr + (size_t)NN * HCC;      // E*H
    unsigned* amax_u  = (unsigned*)(alpha + (size_t)EE * HH);  // N*H
    float*    denom   = (float*)(amax_u + (size_t)NN * HH);    // N*H
    float*    out_pre = denom + (size_t)NN * HH;    // N*HC
    float*    mean    = out_pre + (size_t)NN * HCC; // HC
    float*    var     = mean + HCC;                 // HC

    // 1) init accumulators
    {
        int total = NN * HCC;
        k_init<<<(total + 255) / 256, 256, 0, stream>>>(out_pre, denom, amax_u);
    }
    // 2) WMMA dual GEMM
    {
        int mtiles = (NN + 15) / 16;    // 3125
        k_gemm_xl_xr<<<mtiles, 256, 0, stream>>>(x, W_l, b_l, W_r, b_r, xl, xr);
    }
    // 3) attention logits + segment max
    {
        long long waves = (long long)EE * HH;              // one wave per (e,h)
        int blocks = (int)((waves * 32 + 255) / 256);      // 400000
        k_edge_logits<<<blocks, 256, 0, stream>>>(src, dst, edge_attr, W_e, att,
                                                  xl, xr, alpha, amax_u);
    }
    // 4) exp + denominator
    {
        int total = EE * HH;
        k_softmax_num<<<(total + 255) / 256, 256, 0, stream>>>(dst, amax_u, alpha, denom);
    }
    // 5) normalize attention weights (cheap pass; removes divides from hot kernel)
    {
        int total = EE * HH;
        k_norm_alpha<<<(total + 255) / 256, 256, 0, stream>>>(dst, denom, alpha);
    }
    // 6) weighted scatter-add
    {
        long long total = (long long)EE * HCC;             // 102.4M
        int blocks = (int)((total + 255) / 256);
        k_aggregate<<<blocks, 256, 0, stream>>>(src, dst, xl, alpha, out_pre);
    }
    // 7) BN stats (one block per channel)
    k_bn_stats<<<HCC, 256, 0, stream>>>(out_pre, bias, mean, var);
    // 8) BN affine + ELU
    {
        int total = NN * HCC;
        k_bn_elu<<<(total + 255) / 256, 256, 0, stream>>>(out_pre, bias, mean, var,
                                                          gamma, beta, out);
    }
}